// SpatialHighDimFilter_22814866277098
// MI455X (gfx1250) — compile-verified
//
#include <hip/hip_runtime.h>

// Spatial high-dim filter: splat (block-sum) -> weird convn blur -> bilinear slice.
// H=1024 W=2048 C=16 sigma=16 pad=2 -> grid 68x132x16.
#define H_  1024
#define W_  2048
#define C_  16
#define SH_ 68
#define SW_ 132

typedef float v2f __attribute__((ext_vector_type(2)));
typedef float v8f __attribute__((ext_vector_type(8)));

// ---------------- Splat: one workgroup per grid cell (block-sum downsample) ------------
// Cell (cy,cx) sums pixels y in [16(cy-2)-8, +15], x in [16(cx-2)-8, +15] (clipped).
// Cells outside the covered range sum nothing and write 0 (matches segment_sum zero init).
__global__ __launch_bounds__(256) void splat_kernel(const float* __restrict__ inp,
                                                    float* __restrict__ G)
{
    const int cx  = blockIdx.x;       // 0..SW_-1
    const int cy  = blockIdx.y;       // 0..SH_-1
    const int tid = threadIdx.x;      // 0..255
    const int py  = tid >> 4, px = tid & 15;
    const int y   = 16 * (cy - 2) - 8 + py;
    const int x   = 16 * (cx - 2) - 8 + px;

    float4 v0 = make_float4(0.f, 0.f, 0.f, 0.f);
    float4 v1 = v0, v2 = v0, v3 = v0;
    if ((unsigned)y < (unsigned)H_ && (unsigned)x < (unsigned)W_) {
        const float4* p = (const float4*)(inp + ((size_t)y * W_ + x) * C_);
        v0 = p[0]; v1 = p[1]; v2 = p[2]; v3 = p[3];
    }

    __shared__ float4 red[256 * 4];           // 16 KB
    red[tid * 4 + 0] = v0; red[tid * 4 + 1] = v1;
    red[tid * 4 + 2] = v2; red[tid * 4 + 3] = v3;
    __syncthreads();

    for (int s = 128; s > 0; s >>= 1) {
        if (tid < s) {
#pragma unroll
            for (int q = 0; q < 4; ++q) {
                float4 a = red[tid * 4 + q];
                float4 b = red[(tid + s) * 4 + q];
                a.x += b.x; a.y += b.y; a.z += b.z; a.w += b.w;
                red[tid * 4 + q] = a;
            }
        }
        __syncthreads();
    }
    if (tid < C_)
        G[((size_t)cy * SW_ + cx) * C_ + tid] = ((const float*)red)[tid];
}

// ---------------- Blur pass 1 (iteration 1 of convn, faithful semantics) --------------
// A[y,x] = interior x : (G[y,x-1]+G[y,x+1]+2*G[y,x])/4
//          boundary x : interior y ? (G[y-1,x]+G[y+1,x]+2*G[y,x])/4 : 0
__global__ __launch_bounds__(256) void blurA_kernel(const float* __restrict__ G,
                                                    float* __restrict__ A)
{
    const int idx  = blockIdx.x * 256 + threadIdx.x;   // exact: SH_*SW_*C_ threads
    const int cell = idx >> 4;
    const int x    = cell % SW_;
    const int y    = cell / SW_;
    const float gc = G[idx];
    float r;
    if (x > 0 && x < SW_ - 1)
        r = (G[idx - C_] + G[idx + C_] + 2.0f * gc) * 0.25f;
    else if (y > 0 && y < SH_ - 1)
        r = (G[idx - C_ * SW_] + G[idx + C_ * SW_] + 2.0f * gc) * 0.25f;
    else
        r = 0.0f;
    A[idx] = r;
}

// ---------------- Blur pass 2 (iteration 2 of convn) ----------------------------------
// B[y,x] = interior x : (A[y,x-1]+A[y,x+1]+2*A[y,x])/4
//          boundary x : interior y ? (A[y-1,x]+A[y+1,x]+2*A[y,x])/4 : G[y,x]
__global__ __launch_bounds__(256) void blurB_kernel(const float* __restrict__ G,
                                                    const float* __restrict__ A,
                                                    float* __restrict__ B)
{
    const int idx  = blockIdx.x * 256 + threadIdx.x;
    const int cell = idx >> 4;
    const int x    = cell % SW_;
    const int y    = cell / SW_;
    const float ac = A[idx];
    float r;
    if (x > 0 && x < SW_ - 1)
        r = (A[idx - C_] + A[idx + C_] + 2.0f * ac) * 0.25f;
    else if (y > 0 && y < SH_ - 1)
        r = (A[idx - C_ * SW_] + A[idx + C_ * SW_] + 2.0f * ac) * 0.25f;
    else
        r = G[idx];
    B[idx] = r;
}

// ---------------- Slice via WMMA -------------------------------------------------------
// Each 16x16 output tile (k,j) uses one bilinear quad: cells (k+2..k+3, j+2..j+3).
// Out(256x16) = W(256x4) x Gquad(4x16). One wave per tile, 16x v_wmma_f32_16x16x4_f32.
// A (16x4 f32): lanes 0-15 hold M=m rows with K={0,1}; lanes 16-31 hold K={2,3}.
// B (4x16 f32): VGPR0 = rows K0(l0-15)/K2(l16-31), VGPR1 = rows K1/K3.
// K mapping: 0:(dy0,dx0) 1:(dy0,dx1) 2:(dy1,dx0) 3:(dy1,dx1).
__global__ __launch_bounds__(256) void slice_kernel(const float* __restrict__ Bg,
                                                    float* __restrict__ out)
{
    const int wave = threadIdx.x >> 5;
    const int lane = threadIdx.x & 31;
    const int tile = blockIdx.x * 8 + wave;     // 0..8191
    const int k    = tile >> 7;                 // tile row 0..63
    const int j    = tile & 127;                // tile col 0..127
    const int m    = lane & 15;                 // = px for A rows, = channel for B/D
    const int hi   = lane >> 4;

    // B operand: rows dy = hi, columns = channel m
    const float* gB = Bg + (((size_t)(k + 2 + hi) * SW_) + (j + 2)) * C_ + m;
    v2f b;
    b.x = gB[0];      // (dy=hi, dx=0)
    b.y = gB[C_];     // (dy=hi, dx=1)

    const float am = (float)m * 0.0625f;        // ax for column m
    const float w0 = 1.0f - am;

    float* obase = out + ((size_t)(16 * k) * W_ + 16 * j) * C_;

#pragma unroll
    for (int py = 0; py < 16; ++py) {
        const float ay = (float)py * 0.0625f;
        const float wy = hi ? ay : (1.0f - ay);
        v2f a;
        a.x = wy * w0;   // K = 2*hi + 0
        a.y = wy * am;   // K = 2*hi + 1
        v8f cacc = {};
        v8f d = __builtin_amdgcn_wmma_f32_16x16x4_f32(
            /*neg_a=*/false, a, /*neg_b=*/false, b,
            /*c_mod=*/(short)0, cacc, /*reuse_a=*/false, /*reuse_b=*/false);

        // D row r -> pixel px = r + 8*hi, channel = m
        float* orow = obase + (size_t)py * W_ * C_ + (size_t)hi * 8 * C_ + m;
#pragma unroll
        for (int r = 0; r < 8; ++r)
            orow[r * C_] = d[r];
    }
}

extern "C" void kernel_launch(void* const* d_in, const int* in_sizes, int n_in,
                              void* d_out, int out_size, void* d_ws, size_t ws_size,
                              hipStream_t stream)
{
    const float* inp = (const float*)d_in[0];
    float* out = (float*)d_out;

    const size_t S = (size_t)SH_ * SW_ * C_;   // 143616 floats per grid buffer
    float* G  = (float*)d_ws;
    float* A  = G + S;
    float* Bg = A + S;

    // 1) splat: one block per grid cell (full 68x132 coverage -> implicit zero init)
    splat_kernel<<<dim3(SW_, SH_), 256, 0, stream>>>(inp, G);

    // 2) blur (two dependent elementwise passes over the tiny grid)
    const int n = SH_ * SW_ * C_;              // 143616 = 561 * 256 exactly
    blurA_kernel<<<n / 256, 256, 0, stream>>>(G, A);
    blurB_kernel<<<n / 256, 256, 0, stream>>>(G, A, Bg);

    // 3) slice: 8192 tiles, 1 wave per tile, 8 waves per block
    slice_kernel<<<8192 / 8, 256, 0, stream>>>(Bg, out);
}